// AtomValuator_12077448037108
// MI455X (gfx1250) — compile-verified
//
#include <hip/hip_runtime.h>
#include <hip/hip_bf16.h>

// CDNA5 / gfx1250 fused relational-MLP evaluator (round 3).
//  - f32 WMMA 16x16x4, all layers fused per 64-row tile in LDS.
//  - Transposed W panel in LDS, stride 18 (8B aligned, conflict-free):
//    every B-fragment = one ds_load_b64 with immediate offset, no VALU addr.
//  - Explicit depth-1 software pipeline: frags for ks+4 issued before the
//    WMMA batch for ks -> dscnt waits overlap with WMMA execution.
//  - No lambdas (round 2 spilled the capture frame to scratch).

typedef __attribute__((ext_vector_type(2))) float v2f;
typedef __attribute__((ext_vector_type(4))) float v4f;
typedef __attribute__((ext_vector_type(8))) float v8f;

__device__ __forceinline__ float mish_f(float x) {
    float sp = (x > 20.0f) ? x : log1pf(__expf(x));
    return x * tanhf(sp);
}

// -------- pooling: pooled[s][f] = sum_i E[s*64+i][f] ------------------------
__global__ __launch_bounds__(64) void pool_kernel(const float* __restrict__ E,
                                                  float* __restrict__ pooled) {
    const int s = blockIdx.x;
    const int f = threadIdx.x;
    float acc = 0.0f;
    #pragma unroll 8
    for (int i = 0; i < 64; ++i) acc += E[(s * 64 + i) * 64 + f];
    pooled[s * 64 + f] = acc;
}

// -------- one fused GEMM layer: h <- mish(h[:, :Krows] @ Wsrc + bias) -------
template <int D>
__device__ __forceinline__ void gemm_layer(
    float (&h)[64][16 * D + 4], float (&Wt)[16 * D][18],
    const float* __restrict__ cvecS, const float* __restrict__ Wsrc,
    const int Krows, const float* __restrict__ gbias, const bool useCvec,
    const int tid, const int rowBase, const int wr, const int kh,
    const int rOff) {
    constexpr int d = 16 * D;

    v8f acc[D];
    #pragma unroll
    for (int t = 0; t < D; ++t)
        #pragma unroll
        for (int e = 0; e < 8; ++e) acc[t][e] = 0.0f;

    // v2f-typed LDS bases: all fragment loads become ds_load_b64 + imm offset.
    const v2f* __restrict__ aBase = (const v2f*)(&h[rowBase + wr][kh]);
    const v2f* __restrict__ bBase = (const v2f*)(&Wt[wr][kh]);

    for (int k0 = 0; k0 < Krows; k0 += 16) {
        __syncthreads();  // prev panel fully consumed / h visible
        // stage transposed W panel: Wt[c][kk] = W[k0+kk][c]  (b128 reads)
        for (int idx = tid; idx < 16 * (d / 4); idx += 128) {
            const int kk = idx / (d / 4);
            const int c4 = (idx - kk * (d / 4)) * 4;
            v4f w = *(const v4f*)(Wsrc + (k0 + kk) * d + c4);
            #pragma unroll
            for (int e = 0; e < 4; ++e) Wt[c4 + e][kk] = w[e];
        }
        __syncthreads();

        // depth-1 software-pipelined fragment loads
        v2f aF[2];
        v2f bF[2][D];
        aF[0] = aBase[(k0 + 0) >> 1];
        #pragma unroll
        for (int t = 0; t < D; ++t) bF[0][t] = bBase[t * 144];
        #pragma unroll
        for (int ks = 0; ks < 16; ks += 4) {
            const int cur = (ks >> 2) & 1, nxt = cur ^ 1;
            if (ks + 4 < 16) {  // prefetch next k-step's fragments
                aF[nxt] = aBase[(k0 + ks + 4) >> 1];
                #pragma unroll
                for (int t = 0; t < D; ++t)
                    bF[nxt][t] = bBase[t * 144 + ((ks + 4) >> 1)];
            }
            #pragma unroll
            for (int t = 0; t < D; ++t)
                acc[t] = __builtin_amdgcn_wmma_f32_16x16x4_f32(
                    false, aF[cur], false, bF[cur][t], (short)0, acc[t], false,
                    false);
        }
    }
    __syncthreads();  // all waves done reading h
    #pragma unroll
    for (int t = 0; t < D; ++t) {
        const int col = t * 16 + wr;
        const float bv = useCvec ? cvecS[col] : gbias[col];
        #pragma unroll
        for (int v = 0; v < 8; ++v)
            h[rowBase + rOff + v][col] = mish_f(acc[t][v] + bv);
    }
    __syncthreads();
}

// -------- fused 3-layer MLP + output projection -----------------------------
// ARITY=2: grid 4096, wg=(s,i), rows j=0..63, d=192 (WMMA K: 64,192,192)
// ARITY=1: grid 64,   wg=s,     rows i=0..63, d=128 (WMMA K: 64,128,128)
// ARITY=0: grid 1,               rows s=0..63, d=64  (WMMA K: 64,64,64)
template <int ARITY>
__global__ __launch_bounds__(128) void mlp_fused_kernel(
    const float* __restrict__ E, const float* __restrict__ pooled,
    const float* __restrict__ W1, const float* __restrict__ b1,
    const float* __restrict__ W2, const float* __restrict__ b2,
    const float* __restrict__ W3, const float* __restrict__ b3,
    const float* __restrict__ Wo, const float* __restrict__ bo,
    float* __restrict__ out) {
    constexpr int D  = 4 * (ARITY + 1);  // 16-wide column tiles
    constexpr int d  = 16 * D;           // feature width
    constexpr int DP = d + 4;            // (d+4)%64==4 -> conflict-free A-frags
    __shared__ float h[64][DP];          // activation tile
    __shared__ float Wt[d][18];          // transposed W panel (8B-aligned cols)
    __shared__ float cvec[224];          // layer-1 const contribution / scratch

    const int tid     = threadIdx.x;
    const int wave    = tid >> 5;        // 4 waves x 16 rows
    const int lane    = tid & 31;
    const int wg      = blockIdx.x;
    const int wr      = lane & 15;
    const int kh      = (lane >> 4) << 1;  // 0 | 2  (A/B K-half)
    const int rowBase = wave * 16;
    const int rOff    = (lane >> 4) << 3;  // 0 | 8  (C/D row-half)

    // ---- build varying 64-wide input segment into h[.][0:64] ----
    {
        const float* src;
        if (ARITY == 2)      src = E + ((wg >> 6) * 64) * 64;  // rows j: E[s,j]
        else if (ARITY == 1) src = E + (wg * 64) * 64;         // rows i: E[s,i]
        else                 src = pooled;                     // rows s: pooled
        for (int idx = tid; idx < 64 * 16; idx += 128) {
            const int r = idx >> 4, f4 = (idx & 15) * 4;
            v4f v = *(const v4f*)(src + r * 64 + f4);
            #pragma unroll
            for (int e = 0; e < 4; ++e) h[r][f4 + e] = v[e];
        }
    }

    // ---- cvec = b1 + (block-constant input segments) @ W1 ----
    for (int n = tid; n < d; n += 128) {
        float s = b1[n];
        if (ARITY == 2) {
            const int st = wg >> 6, i = wg & 63;
            const float* xi = E + (st * 64 + i) * 64;   // emb_i
            const float* xp = pooled + st * 64;         // pooled_s
            #pragma unroll 4
            for (int f = 0; f < 64; ++f) s += xi[f] * W1[f * d + n];
            #pragma unroll 4
            for (int f = 0; f < 64; ++f) s += xp[f] * W1[(128 + f) * d + n];
        } else if (ARITY == 1) {
            const float* xp = pooled + wg * 64;
            #pragma unroll 4
            for (int f = 0; f < 64; ++f) s += xp[f] * W1[(64 + f) * d + n];
        }
        cvec[n] = s;
    }

    // layer 1 runs WMMA only over the varying 64-wide K segment
    gemm_layer<D>(h, Wt, cvec, ARITY == 2 ? (W1 + 64 * d) : W1, 64, b1, true,
                  tid, rowBase, wr, kh, rOff);
    gemm_layer<D>(h, Wt, cvec, W2, d, b2, false, tid, rowBase, wr, kh, rOff);
    gemm_layer<D>(h, Wt, cvec, W3, d, b3, false, tid, rowBase, wr, kh, rOff);

    // ---- output projection: out[row] = h[row] . Wo + bo (split-K x2) ----
    const int row = tid & 63, half = tid >> 6;
    const int kb = half * (d / 2);
    float s = 0.0f;
    #pragma unroll 4
    for (int k = 0; k < d / 2; ++k) s += h[row][kb + k] * Wo[kb + k];
    if (half) cvec[row] = s;  // reuse cvec as reduction scratch
    __syncthreads();
    if (!half) out[(size_t)wg * 64 + row] = s + cvec[row] + bo[0];
}

extern "C" void kernel_launch(void* const* d_in, const int* in_sizes, int n_in,
                              void* d_out, int out_size, void* d_ws, size_t ws_size,
                              hipStream_t stream) {
    // d_in order: embeddings, batch_info, then 8 tensors per MLP head
    // (W1,b1,W2,b2,W3,b3,Wo,bo) for on(3F), clear(2F), handempty(1F).
    const float* E = (const float*)d_in[0];
    // d_in[1] (batch_info) is implied by the homogeneous layout; unused.
    const float* onP[8];
    const float* clP[8];
    const float* heP[8];
    for (int k = 0; k < 8; ++k) {
        onP[k] = (const float*)d_in[2 + k];
        clP[k] = (const float*)d_in[10 + k];
        heP[k] = (const float*)d_in[18 + k];
    }

    float* out       = (float*)d_out;
    float* out_on    = out;                  // 262144
    float* out_clear = out + 262144;         // 4096
    float* out_he    = out + 262144 + 4096;  // 64
    float* pooled    = (float*)d_ws;         // 64*64 floats scratch

    pool_kernel<<<64, 64, 0, stream>>>(E, pooled);

    mlp_fused_kernel<2><<<4096, 128, 0, stream>>>(
        E, pooled, onP[0], onP[1], onP[2], onP[3], onP[4], onP[5], onP[6], onP[7],
        out_on);
    mlp_fused_kernel<1><<<64, 128, 0, stream>>>(
        E, pooled, clP[0], clP[1], clP[2], clP[3], clP[4], clP[5], clP[6], clP[7],
        out_clear);
    mlp_fused_kernel<0><<<1, 128, 0, stream>>>(
        E, pooled, heP[0], heP[1], heP[2], heP[3], heP[4], heP[5], heP[6], heP[7],
        out_he);
}